// GreedyRouter_49417893708015
// MI455X (gfx1250) — compile-verified
//
#include <hip/hip_runtime.h>
#include <hip/hip_bf16.h>
#include <stdint.h>

// ---------------------------------------------------------------------------
// MoE greedy router: softmax(64) -> top-8 (sorted, low-index ties) -> renorm
// Memory-bound: ~436MB traffic -> ~19us floor @ 23.3TB/s. No matmul structure
// => WMMA inapplicable; CDNA5 path exercised = async global<->LDS movers
// (ASYNCcnt) for the coalesced tile staging, wave32-safe per-thread top-k.
// ---------------------------------------------------------------------------

#define AS1 __attribute__((address_space(1)))
#define AS3 __attribute__((address_space(3)))

// Builtin pointer element type discovered from hipcc diagnostic:
// 'int __attribute__((__vector_size__(4 * sizeof(int)))) __device__ *'
typedef int router_v4i __attribute__((vector_size(16)));
typedef AS1 router_v4i* router_gptr4;  // global (AS1 / "__device__") int4*
typedef AS3 router_v4i* router_lptr4;  // LDS (AS3 / "__shared__") int4*

#define BLOCK_THREADS 128
#define TOK_PER_BLOCK 128
#define NEXP 64
#define TOPK 8
#define NVEC 16                               // float4 per token row
#define ROW_STRIDE 68                         // floats; 272B rows: 16B-aligned, <=2-way bank conflict
#define VEC_ITERS ((TOK_PER_BLOCK * NVEC) / BLOCK_THREADS)  // 16

#if defined(__gfx1250__) && __has_builtin(__builtin_amdgcn_global_load_async_to_lds_b128)
#define HAVE_ASYNC_LOAD 1
#else
#define HAVE_ASYNC_LOAD 0
#endif
#if defined(__gfx1250__) && __has_builtin(__builtin_amdgcn_global_store_async_from_lds_b128)
#define HAVE_ASYNC_STORE 1
#else
#define HAVE_ASYNC_STORE 0
#endif

static __device__ __forceinline__ void router_wait_async0() {
#if defined(__gfx1250__)
#if __has_builtin(__builtin_amdgcn_s_wait_asynccnt)
  __builtin_amdgcn_s_wait_asynccnt(0);
#else
  asm volatile("s_wait_asynccnt 0" ::: "memory");
#endif
#endif
}

// Insert packed key into descending sorted 8-list (pure v_max_u32/v_min_u32).
static __device__ __forceinline__ void ins8(unsigned (&s)[8], float val, unsigned idx) {
  // probs > 0 => IEEE bits are order-isomorphic as unsigned.
  // Low 6 bits carry (63 - idx): on equal truncated value, LOWER index wins
  // the max — matches jax.lax.top_k / torch.topk stable tie-break.
  unsigned t = (__float_as_uint(val) & 0xFFFFFFC0u) | (63u - idx);
#pragma unroll
  for (int j = 0; j < 8; ++j) {
    const unsigned hi = (s[j] >= t) ? s[j] : t;
    const unsigned lo = (s[j] >= t) ? t : s[j];
    s[j] = hi;
    t = lo;
  }
}

__global__ void moe_router_zero_hist(float* __restrict__ h) {
  if (threadIdx.x < NEXP) h[threadIdx.x] = 0.0f;
}

__global__ __launch_bounds__(BLOCK_THREADS) void
moe_softmax_top8_kernel(const float* __restrict__ logits,
                        float* __restrict__ out_logits,
                        float* __restrict__ out_probs,
                        float* __restrict__ out_topk_w,
                        float* __restrict__ out_topk_id,
                        float* __restrict__ out_hist,
                        int T) {
  __shared__ float tile[TOK_PER_BLOCK * ROW_STRIDE];
  __shared__ unsigned hist[NEXP];

  const int tid = threadIdx.x;
  const long long t0 = (long long)blockIdx.x * TOK_PER_BLOCK;

  if (tid < NEXP) hist[tid] = 0u;

  // -------- Phase 1: coalesced stage of 128x64 f32 tile into LDS (async on
  // CDNA5, bypassing VGPRs) + logits passthrough copy. -----------------------
  const float4* gin = (const float4*)logits + t0 * NVEC;
  float4* gpass = (float4*)out_logits + t0 * NVEC;
#pragma unroll
  for (int j = 0; j < VEC_ITERS; ++j) {
    const int v = tid + j * BLOCK_THREADS;  // float4 index inside tile
    const int tokLocal = v >> 4;
    const int vcol = v & 15;
    if (t0 + tokLocal < T) {
      float* lptr = &tile[tokLocal * ROW_STRIDE + vcol * 4];
#if HAVE_ASYNC_LOAD
      __builtin_amdgcn_global_load_async_to_lds_b128(
          (router_gptr4)(gin + v), (router_lptr4)lptr, 0, 0);
      gpass[v] = gin[v];  // passthrough (L0/L2 hit)
#else
      const float4 x = gin[v];
      gpass[v] = x;
      *(float4*)lptr = x;
#endif
    }
  }
#if HAVE_ASYNC_LOAD
  router_wait_async0();
#endif
  __syncthreads();

  // -------- Phase 2: one thread per token --------------------------------
  const long long tok = t0 + tid;
  if (tok < T) {
    float4* row = (float4*)&tile[tid * ROW_STRIDE];
    float4 r[NVEC];
#pragma unroll
    for (int j = 0; j < NVEC; ++j) r[j] = row[j];  // ds_load_b128 x16

    float m = r[0].x;
#pragma unroll
    for (int j = 0; j < NVEC; ++j)
      m = fmaxf(m, fmaxf(fmaxf(r[j].x, r[j].y), fmaxf(r[j].z, r[j].w)));

    float sum = 0.0f;
#pragma unroll
    for (int j = 0; j < NVEC; ++j) {
      r[j].x = __expf(r[j].x - m);
      r[j].y = __expf(r[j].y - m);
      r[j].z = __expf(r[j].z - m);
      r[j].w = __expf(r[j].w - m);
      sum += r[j].x + r[j].y + r[j].z + r[j].w;
    }
    const float inv = 1.0f / sum;

    unsigned s[8] = {0u, 0u, 0u, 0u, 0u, 0u, 0u, 0u};
#pragma unroll
    for (int j = 0; j < NVEC; ++j) {
      r[j].x *= inv; r[j].y *= inv; r[j].z *= inv; r[j].w *= inv;
      row[j] = r[j];  // normalized probs back to LDS for phase 3 + refetch
      ins8(s, r[j].x, 4u * j + 0u);
      ins8(s, r[j].y, 4u * j + 1u);
      ins8(s, r[j].z, 4u * j + 2u);
      ins8(s, r[j].w, 4u * j + 3u);
    }

    float pv[TOPK];
    float fid[TOPK];
    float sum8 = 0.0f;
#pragma unroll
    for (int j = 0; j < TOPK; ++j) {
      const int id = 63 - (int)(s[j] & 63u);
      const float p = tile[tid * ROW_STRIDE + id];  // exact (un-truncated) prob
      pv[j] = p;
      fid[j] = (float)id;
      sum8 += p;
      atomicAdd(&hist[id], 1u);
    }
    const float inv8 = 1.0f / sum8;  // renorm (scaling==1.0)

    float4* wdst = (float4*)(out_topk_w + tok * TOPK);
    wdst[0] = make_float4(pv[0] * inv8, pv[1] * inv8, pv[2] * inv8, pv[3] * inv8);
    wdst[1] = make_float4(pv[4] * inv8, pv[5] * inv8, pv[6] * inv8, pv[7] * inv8);
    float4* idst = (float4*)(out_topk_id + tok * TOPK);
    idst[0] = make_float4(fid[0], fid[1], fid[2], fid[3]);
    idst[1] = make_float4(fid[4], fid[5], fid[6], fid[7]);
  }
  __syncthreads();

  // -------- Phase 3: coalesced stream-out of routing_weights from LDS -----
  float4* gprob = (float4*)out_probs + t0 * NVEC;
#pragma unroll
  for (int j = 0; j < VEC_ITERS; ++j) {
    const int v = tid + j * BLOCK_THREADS;
    const int tokLocal = v >> 4;
    const int vcol = v & 15;
    if (t0 + tokLocal < T) {
      float* lptr = &tile[tokLocal * ROW_STRIDE + vcol * 4];
#if HAVE_ASYNC_STORE
      __builtin_amdgcn_global_store_async_from_lds_b128(
          (router_gptr4)(gprob + v), (router_lptr4)lptr, 0, 0);
#else
      gprob[v] = *(const float4*)lptr;
#endif
    }
  }

  // Block-level histogram flush: integer counts summed in f32 are exact and
  // order-independent => deterministic despite atomics.
  if (tid < NEXP) {
    const unsigned c = hist[tid];
    if (c) atomicAdd(&out_hist[tid], (float)c);
  }
#if HAVE_ASYNC_STORE
  router_wait_async0();  // LDS must stay live until async stores drain
#endif
}

extern "C" void kernel_launch(void* const* d_in, const int* in_sizes, int n_in,
                              void* d_out, int out_size, void* d_ws, size_t ws_size,
                              hipStream_t stream) {
  const float* logits = (const float*)d_in[0];
  const long long total = in_sizes[0];
  const int T = (int)(total / NEXP);

  float* out0 = (float*)d_out;                    // logits passthrough  [T,64]
  float* out1 = out0 + (long long)T * NEXP;       // routing_weights     [T,64]
  float* out2 = out1 + (long long)T * NEXP;       // topk_weights        [T,8]
  float* out3 = out2 + (long long)T * TOPK;       // topk_ids (as f32)   [T,8]
  float* out4 = out3 + (long long)T * TOPK;       // tokens_per_expert   [64]

  moe_router_zero_hist<<<1, NEXP, 0, stream>>>(out4);

  const int blocks = (T + TOK_PER_BLOCK - 1) / TOK_PER_BLOCK;
  moe_softmax_top8_kernel<<<blocks, BLOCK_THREADS, 0, stream>>>(
      logits, out0, out1, out2, out3, out4, T);
}